// ClassLoss_54382875902459
// MI455X (gfx1250) — compile-verified
//
#include <hip/hip_runtime.h>
#include <hip/hip_bf16.h>

// B=32, A=3, H=64, W=64, NC=80, M=50
#define BB 32
#define AA 3
#define HH 64
#define WW 64
#define NC 80
#define MM 50
#define CH 85            // 5 + NC channels in output
#define CELLS (HH * WW)  // 4096
#define ROWS_PER_B (AA * HH * WW)  // 12288
#define LIST_CAP 64

typedef __attribute__((ext_vector_type(2))) float v2f;
typedef __attribute__((ext_vector_type(8))) float v8f;

__global__ void zero_out_kernel(float* out) { out[0] = 0.0f; }

// One block per batch: build gt map in LDS, scatter targets (single thread,
// deterministic overwrite order), then wave-0 ballot-compacts active cells
// into ws as (cell | label<<16).
__global__ void build_compact_kernel(const float* __restrict__ targets,
                                     int* __restrict__ list,
                                     int* __restrict__ nactive) {
    __shared__ int gt[CELLS];
    const int b = blockIdx.x;
    const int t = threadIdx.x;

    for (int i = t; i < CELLS; i += blockDim.x) gt[i] = -100;
    __syncthreads();

    if (t == 0) {
        const float* tb = targets + (size_t)b * MM * 5;
        for (int m = 0; m < MM; ++m) {
            const float c0 = tb[m * 5 + 0];
            const float x  = tb[m * 5 + 1];
            const float y  = tb[m * 5 + 2];
            const float w  = tb[m * 5 + 3];
            const float h  = tb[m * 5 + 4];
            const bool valid = !(c0 == 0.0f && x == 0.0f && y == 0.0f &&
                                 w == 0.0f && h == 0.0f);
            if (!valid) continue;  // reference sets row=H -> dropped
            const int r = (int)(y * (float)HH);
            const int c = (int)(x * (float)WW);
            if (r >= 0 && r < HH && c >= 0 && c < WW)
                gt[r * WW + c] = (int)c0;  // later targets overwrite earlier
        }
    }
    __syncthreads();

    if (t < 32) {
        const int lane = t;
        int cnt = 0;
        for (int base = 0; base < CELLS; base += 32) {
            const int cell = base + lane;
            const int lab = gt[cell];
            const unsigned bal = __builtin_amdgcn_ballot_w32(lab >= 0);
            if (lab >= 0) {
                const int pos = cnt + __popc(bal & ((1u << lane) - 1u));
                if (pos < LIST_CAP)
                    list[b * LIST_CAP + pos] = cell | (lab << 16);
            }
            cnt += __popc(bal);
        }
        if (lane == 0) nactive[b] = cnt < LIST_CAP ? cnt : LIST_CAP;
    }
}

// One wave (32 lanes) per batch. Processes active rows 16 at a time:
//   pass 1: per-lane row max + logit at label (lanes L and L+16 redundantly
//           compute row (g*16 + L%16) so no cross-lane broadcast is needed)
//   pass 2: 20x V_WMMA_F32_16X16X4_F32 with B = ones accumulates
//           sum_k exp(x_k - max) per row in full f32.
__global__ void loss_kernel(const float* __restrict__ output,
                            const int* __restrict__ list,
                            const int* __restrict__ nactive,
                            float* __restrict__ d_out) {
    const int b = blockIdx.x;
    const int L = threadIdx.x;  // 0..31
    __shared__ float sums[16];

    const int ncell = nactive[b];
    const int nrows = ncell * AA;
    const int ngroups = (nrows + 15) >> 4;
    const float* batch_base = output + (size_t)b * ROWS_PER_B * CH;

    float lsum = 0.0f;
    float lcnt = 0.0f;

    v2f bones;
    bones[0] = 1.0f;
    bones[1] = 1.0f;
    const int koff = (L >> 4) << 1;  // A-matrix K offset for this half-wave

    for (int g = 0; g < ngroups; ++g) {
        const int mr = g * 16 + (L & 15);
        const bool act = (mr < nrows) && (ncell > 0);
        const int mrc = act ? mr : 0;  // inactive lanes alias row 0 (valid mem)

        const int entry = list[b * LIST_CAP + mrc / AA];
        const int cell = entry & 0xFFFF;
        const int label = entry >> 16;
        const int a = mrc % AA;
        // flat row index i = cell*A + a matches BOTH the (H,W,A) label
        // flattening and the i-th contiguous 80-chunk of output[b][...,5:]
        const float* row = batch_base + (size_t)(cell * AA + a) * CH + 5;

        // pass 1: row max and label logit
        float mx = row[0];
        #pragma unroll 4
        for (int k = 1; k < NC; ++k) mx = fmaxf(mx, row[k]);
        const float xlab = row[label];

        // pass 2: WMMA sum of exp(x - mx) over 80 classes (20 chunks of K=4)
        v8f c = {};
        for (int ch = 0; ch < NC / 4; ++ch) {
            const int k0 = ch * 4 + koff;
            v2f a2;
            a2[0] = __expf(row[k0] - mx);
            a2[1] = __expf(row[k0 + 1] - mx);
            c = __builtin_amdgcn_wmma_f32_16x16x4_f32(
                false, a2, false, bones, (short)0, c, false, false);
        }

        // D layout: lane 0 VGPR r = rowsum(r); lane 16 VGPR r = rowsum(r+8)
        if (L == 0) {
            #pragma unroll
            for (int r = 0; r < 8; ++r) sums[r] = c[r];
        }
        if (L == 16) {
            #pragma unroll
            for (int r = 0; r < 8; ++r) sums[8 + r] = c[r];
        }
        __syncthreads();
        if (L < 16 && act) {
            const float rs = sums[L];
            lsum += -(xlab - mx - logf(rs));
            lcnt += 1.0f;
        }
        __syncthreads();
    }

    // wave32 reduction
    for (int off = 16; off > 0; off >>= 1) {
        lsum += __shfl_down(lsum, off, 32);
        lcnt += __shfl_down(lcnt, off, 32);
    }
    if (L == 0) {
        const float loss_b = lsum / fmaxf(lcnt, 1.0f);
        atomicAdd(d_out, loss_b * (1.0f / (float)BB));
    }
}

extern "C" void kernel_launch(void* const* d_in, const int* in_sizes, int n_in,
                              void* d_out, int out_size, void* d_ws, size_t ws_size,
                              hipStream_t stream) {
    (void)in_sizes; (void)n_in; (void)out_size; (void)ws_size;
    const float* output  = (const float*)d_in[0];
    const float* targets = (const float*)d_in[1];
    float* out = (float*)d_out;

    int* list    = (int*)d_ws;                 // BB * LIST_CAP ints
    int* nactive = list + BB * LIST_CAP;       // BB ints

    zero_out_kernel<<<1, 1, 0, stream>>>(out);
    build_compact_kernel<<<BB, 256, 0, stream>>>(targets, list, nactive);
    loss_kernel<<<BB, 32, 0, stream>>>(output, list, nactive, out);
}